// PSELoss_75428215653181
// MI455X (gfx1250) — compile-verified
//
#include <hip/hip_runtime.h>

#define B_    16
#define C_    7
#define K_    6
#define Hc    160
#define Wc    160
#define Hf    640
#define Wf    640
#define Nf    (Hf*Wf)          // 409600 pixels per batch image
#define EPSf  1e-6f
#define ALPHAf 0.7f
#define RATIO 3

// ---- workspace layout, in 4-byte units ----
#define WS_HIST  0u                     // u32[16*65536] (reused hi then lo pass)
#define WS_POS   (16u*65536u)           // u32[16]
#define WS_NEGT  (WS_POS+16u)           // u32[16]
#define WS_BSTAR (WS_NEGT+16u)          // u32[16]
#define WS_RANK  (WS_BSTAR+16u)         // u32[16]
#define WS_FALL  (WS_RANK+16u)          // u32[16]
#define WS_TH    (WS_FALL+16u)          // f32[16]
#define WS_ACC   (WS_TH+16u)            // f32[16*32] (29 used per batch)
#define WS_TOTAL (WS_ACC+16u*32u)

typedef unsigned int u32x4_t __attribute__((ext_vector_type(4)));
typedef int          i32x4_t __attribute__((ext_vector_type(4)));
typedef int          i32x8_t __attribute__((ext_vector_type(8)));

__device__ __forceinline__ unsigned sortkey(float f){
  unsigned u = __float_as_uint(f);
  return u ^ ((u >> 31) ? 0xFFFFFFFFu : 0x80000000u);   // monotone: float order -> uint order
}
__device__ __forceinline__ float key2float(unsigned k){
  unsigned u = (k & 0x80000000u) ? (k ^ 0x80000000u) : ~k;
  return __uint_as_float(u);
}
__device__ __forceinline__ float sigmoidf_(float x){
  return 1.0f / (1.0f + __expf(-x));
}

// -------------------- zero workspace --------------------
__global__ void kz(unsigned* __restrict__ ws){
  unsigned i = blockIdx.x * 256u + threadIdx.x;
  if (i < WS_TOTAL) ws[i] = 0u;
}

// -------------------- pass 1: hi-16 histogram of neg scores + pos/neg counts --------------------
__global__ void k0_hist_hi(const float* __restrict__ maps,
                           const float* __restrict__ gt,
                           const float* __restrict__ tm,
                           unsigned* __restrict__ ws){
  const int yc = blockIdx.x, b = blockIdx.y;
  const int tid = threadIdx.x;
  int posc = 0, negt = 0;
  if (tid < Wc){
    const float s = maps[((b*C_ + 0)*Hc + yc)*Wc + tid];
    const unsigned hi = sortkey(s) >> 16;
    int negc = 0;
    for (int fy = 0; fy < 4; ++fy){
      const int base = (b*Hf + yc*4 + fy)*Wf + tid*4;
      const float4 g = *reinterpret_cast<const float4*>(gt + base);
      const float4 m = *reinterpret_cast<const float4*>(tm + base);
      const float gv[4] = {g.x, g.y, g.z, g.w};
      const float mv[4] = {m.x, m.y, m.z, m.w};
      for (int j = 0; j < 4; ++j){
        const bool pos = gv[j] > 0.5f;
        negc += pos ? 0 : 1;
        posc += (pos && mv[j] > 0.5f) ? 1 : 0;
      }
    }
    negt = negc;
    if (negc) atomicAdd(&ws[WS_HIST + (unsigned)b*65536u + hi], (unsigned)negc);
  }
  for (int off = 16; off; off >>= 1){
    posc += __shfl_xor(posc, off, 32);
    negt += __shfl_xor(negt, off, 32);
  }
  if ((tid & 31) == 0){
    atomicAdd(&ws[WS_POS  + b], (unsigned)posc);
    atomicAdd(&ws[WS_NEGT + b], (unsigned)negt);
  }
}

// -------------------- select hi bucket containing rank R; then clear hist --------------------
__global__ void k1_select_hi(unsigned* __restrict__ ws){
  const int b = blockIdx.x, tid = threadIdx.x;
  unsigned* h = ws + WS_HIST + (unsigned)b*65536u;
  __shared__ unsigned csum[256];
  const int cbase = (255 - tid)*256;              // chunk order: top-down
  unsigned s = 0;
  for (int j = 0; j < 256; ++j) s += h[cbase + j];
  csum[tid] = s;
  __syncthreads();
  if (tid == 0){
    const unsigned pos  = ws[WS_POS  + b];
    const unsigned negt = ws[WS_NEGT + b];
    unsigned long long nn = (unsigned long long)pos * (unsigned long long)RATIO;
    const unsigned neg_num = (unsigned)(nn < (unsigned long long)negt ? nn : (unsigned long long)negt);
    const unsigned fall = (pos == 0u || neg_num == 0u) ? 1u : 0u;
    long long idx = (long long)neg_num - 1;
    if (idx < 0) idx = 0;
    if (idx > (long long)(Nf - 1)) idx = Nf - 1;
    const unsigned R = (unsigned)idx + 1u;
    unsigned cum = 0, rank = 0, bstar = 0;
    if (!fall){
      for (int c = 0; c < 256; ++c){
        if (cum + csum[c] >= R){
          const int top = 65535 - c*256;
          for (int bkt = top; bkt >= top - 255; --bkt){
            cum += h[bkt];
            if (cum >= R){ bstar = (unsigned)bkt; rank = R - (cum - h[bkt]); break; }
          }
          break;
        }
        cum += csum[c];
      }
    }
    ws[WS_BSTAR + b] = bstar;
    ws[WS_RANK  + b] = rank;
    ws[WS_FALL  + b] = fall;
  }
  __syncthreads();
  for (int j = 0; j < 256; ++j) h[cbase + j] = 0u;   // reuse buffer for lo pass
}

// -------------------- pass 2: lo-16 histogram restricted to hi bucket --------------------
__global__ void k2_hist_lo(const float* __restrict__ maps,
                           const float* __restrict__ gt,
                           unsigned* __restrict__ ws){
  const int yc = blockIdx.x, b = blockIdx.y;
  const int tid = threadIdx.x;
  if (ws[WS_FALL + b]) return;
  if (tid >= Wc) return;
  const unsigned bstar = ws[WS_BSTAR + b];
  const float s = maps[((b*C_ + 0)*Hc + yc)*Wc + tid];
  const unsigned key = sortkey(s);
  if ((key >> 16) != bstar) return;
  int negc = 0;
  for (int fy = 0; fy < 4; ++fy){
    const int base = (b*Hf + yc*4 + fy)*Wf + tid*4;
    const float4 g = *reinterpret_cast<const float4*>(gt + base);
    negc += (g.x <= 0.5f) + (g.y <= 0.5f) + (g.z <= 0.5f) + (g.w <= 0.5f);
  }
  if (negc) atomicAdd(&ws[WS_HIST + (unsigned)b*65536u + (key & 0xFFFFu)], (unsigned)negc);
}

// -------------------- recover exact float threshold --------------------
__global__ void k3_select_lo(unsigned* __restrict__ ws){
  const int b = blockIdx.x, tid = threadIdx.x;
  unsigned* h = ws + WS_HIST + (unsigned)b*65536u;
  __shared__ unsigned csum[256];
  const int cbase = (255 - tid)*256;
  unsigned s = 0;
  for (int j = 0; j < 256; ++j) s += h[cbase + j];
  csum[tid] = s;
  __syncthreads();
  if (tid == 0){
    float th = -__builtin_inff();
    if (!ws[WS_FALL + b]){
      const unsigned R = ws[WS_RANK + b];
      unsigned cum = 0, lo = 0;
      for (int c = 0; c < 256; ++c){
        if (cum + csum[c] >= R){
          const int top = 65535 - c*256;
          for (int bkt = top; bkt >= top - 255; --bkt){
            cum += h[bkt];
            if (cum >= R){ lo = (unsigned)bkt; break; }
          }
          break;
        }
        cum += csum[c];
      }
      th = key2float((ws[WS_BSTAR + b] << 16) | lo);
    }
    ((float*)ws)[WS_TH + b] = th;
  }
}

// -------------------- main fused pass --------------------
struct __align__(16) SmemK4 {
  float tile[K_*4*320];     // gt_kernels tile [k][fy][x], filled by TDM DMA
  float coarse[C_][80];     // coarse logits (text + 6 kernel planes)
  float red[32];
};

__global__ void k4_main(const float* __restrict__ maps,
                        const float* __restrict__ gt,
                        const float* __restrict__ gtk,
                        const float* __restrict__ tm,
                        unsigned* __restrict__ ws){
  __shared__ SmemK4 sm;
  const int h   = blockIdx.x;          // half of row (320 cols)
  const int yc  = blockIdx.y;          // coarse row (4 fine rows)
  const int b   = blockIdx.z;
  const int tid = threadIdx.x;

  // ---- TDM: DMA a 3-D tile (6 planes x 4 rows x 320 cols of f32) of gt_kernels into LDS ----
  if (tid < 32){  // wave 0 only (TDM issues per-wave, EXEC ignored)
    const unsigned lds = (unsigned)(unsigned long long)(&sm.tile[0]);   // low 32 bits = LDS offset
    const unsigned long long ga = (unsigned long long)
        (gtk + (((size_t)b*K_)*Hf + (size_t)yc*4)*Wf + (size_t)h*320);
    u32x4_t g0;
    g0[0] = 1u;                                           // count=1, user descriptor
    g0[1] = lds;                                          // lds_addr
    g0[2] = (unsigned)(ga & 0xFFFFFFFFull);               // global_addr[31:0]
    g0[3] = (unsigned)((ga >> 32) & 0x01FFFFFFull) | (2u << 30);  // ga[56:32] | type=2
    i32x8_t g1;
    g1[0] = (int)(2u << 16);                              // data_size = 4 bytes
    g1[1] = (int)((unsigned)Wf << 16);                    // tensor_dim0[15:0]=640
    g1[2] = (int)((unsigned)Wf << 16);                    // td0 hi(0) | tensor_dim1[15:0]=640
    g1[3] = (int)(320u << 16);                            // td1 hi(0) | tile_dim0=320
    g1[4] = (int)(4u | (6u << 16));                       // tile_dim1=4 | tile_dim2=6
    g1[5] = (int)Wf;                                      // tensor_dim0_stride=640 (lo32)
    g1[6] = (int)(((unsigned)Nf & 0xFFFFu) << 16);        // s0 hi(0) | tensor_dim1_stride[15:0]
    g1[7] = (int)((unsigned)Nf >> 16);                    // tensor_dim1_stride[47:16] (409600)
    i32x4_t g2; g2[0] = K_; g2[1] = 0; g2[2] = 0; g2[3] = 0;   // tensor_dim2=6
    i32x4_t g3; g3[0] = 0;  g3[1] = 0; g3[2] = 0; g3[3] = 0;
#if defined(__clang_major__) && (__clang_major__ >= 23)
    i32x8_t g4; for (int z = 0; z < 8; ++z) g4[z] = 0;
    __builtin_amdgcn_tensor_load_to_lds(g0, g1, g2, g3, g4, 0);
#else
    __builtin_amdgcn_tensor_load_to_lds(g0, g1, g2, g3, 0);
#endif
  }

  // stage coarse logits while DMA is in flight
  for (int i = tid; i < C_*80; i += 256){
    const int c = i / 80, x = i % 80;
    sm.coarse[c][x] = maps[((b*C_ + c)*Hc + yc)*Wc + h*80 + x];
  }
  if (tid < 32) sm.red[tid] = 0.0f;
  if (tid < 32) __builtin_amdgcn_s_wait_tensorcnt(0);
  __syncthreads();

  const float th  = ((const float*)ws)[WS_TH + b];
  const bool fall = ws[WS_FALL + b] != 0u;

  float acc[29];
  for (int i = 0; i < 29; ++i) acc[i] = 0.0f;

  // 320 quads of 4 fine pixels; each quad maps to exactly one coarse pixel
  for (int q = tid; q < 320; q += 256){
    const int fy = q / 80, cx = q % 80;
    const int base = (b*Hf + yc*4 + fy)*Wf + h*320 + cx*4;
    const float4 g = *reinterpret_cast<const float4*>(gt + base);
    const float4 m = *reinterpret_cast<const float4*>(tm + base);
    const float s     = sm.coarse[0][cx];
    const float sig_s = sigmoidf_(s);
    const bool  predT = s > 0.0f;

    float kl[K_], pk[K_];
    float4 gk[K_];
    for (int k = 0; k < K_; ++k){
      kl[k] = sm.coarse[1 + k][cx];
      pk[k] = sigmoidf_(kl[k]);
      gk[k] = *reinterpret_cast<const float4*>(&sm.tile[(k*4 + fy)*320 + cx*4]);
    }
    const float gv[4] = {g.x, g.y, g.z, g.w};
    const float mv[4] = {m.x, m.y, m.z, m.w};
    for (int j = 0; j < 4; ++j){
      const float gj = gv[j], mj = mv[j];
      const bool pos = gj > 0.5f;
      float sel;
      if (fall) sel = mj;
      else      sel = (((s >= th) || pos) && (mj > 0.5f)) ? 1.0f : 0.0f;
      const float pm = sig_s * sel;
      const float tv = gj * sel;
      acc[0] += pm * tv;          // dice text a
      acc[1] += pm * pm;          // dice text b
      acc[2] += tv * tv;          // dice text c
      const bool valid = (mj == 1.0f);
      const bool gi = pos;        // gt is exactly 0/1
      if (valid){
        acc[3] += (!predT && !gi) ? 1.0f : 0.0f;  // inter class0
        acc[4] += (!predT || !gi) ? 1.0f : 0.0f;  // union class0
        acc[5] += ( predT &&  gi) ? 1.0f : 0.0f;  // inter class1
        acc[6] += ( predT ||  gi) ? 1.0f : 0.0f;  // union class1
      }
      const float km = predT ? mj : 0.0f;         // kernel_mask = (sigmoid(s)>0.5)*tm
      for (int k = 0; k < K_; ++k){
        const float gkj = ((const float*)&gk[k])[j];
        const float pkj = pk[k] * km;
        const float tkj = gkj * km;
        acc[7  + k] += pkj * tkj;
        acc[13 + k] += pkj * pkj;
        acc[19 + k] += tkj * tkj;
      }
      const bool valid2 = (mj * gj) == 1.0f;      // tm * gt_texts == 1
      if (valid2){
        const bool predK = kl[K_-1] > 0.0f;
        const bool gki   = (((const float*)&gk[K_-1])[j]) > 0.5f;
        acc[25] += (!predK && !gki) ? 1.0f : 0.0f;
        acc[26] += (!predK || !gki) ? 1.0f : 0.0f;
        acc[27] += ( predK &&  gki) ? 1.0f : 0.0f;
        acc[28] += ( predK ||  gki) ? 1.0f : 0.0f;
      }
    }
  }

  // wave reduce -> LDS combine -> per-batch global accumulators
  for (int i = 0; i < 29; ++i){
    float v = acc[i];
    for (int off = 16; off; off >>= 1) v += __shfl_xor(v, off, 32);
    if ((tid & 31) == 0) atomicAdd(&sm.red[i], v);
  }
  __syncthreads();
  if (tid < 29) atomicAdd(&((float*)ws)[WS_ACC + b*32 + tid], sm.red[tid]);
}

// -------------------- finalize 5 outputs --------------------
__global__ void k5_final(const unsigned* __restrict__ ws, float* __restrict__ out){
  const int b = threadIdx.x;
  float l = 0.f, lt = 0.f, lk = 0.f, it = 0.f, ik = 0.f;
  if (b < B_){
    const float* a = (const float*)ws + WS_ACC + b*32;
    lt = 1.0f - 2.0f*a[0] / ((a[1] + EPSf) + (a[2] + EPSf));
    float sk = 0.f;
    for (int k = 0; k < K_; ++k)
      sk += 1.0f - 2.0f*a[7+k] / ((a[13+k] + EPSf) + (a[19+k] + EPSf));
    lk = sk / (float)K_;
    it = 0.5f * (a[3]/(a[4] + EPSf) + a[5]/(a[6] + EPSf));
    ik = 0.5f * (a[25]/(a[26] + EPSf) + a[27]/(a[28] + EPSf));
    l  = ALPHAf*lt + (1.0f - ALPHAf)*lk;
  }
  for (int off = 16; off; off >>= 1){
    l  += __shfl_xor(l,  off, 32);
    lt += __shfl_xor(lt, off, 32);
    lk += __shfl_xor(lk, off, 32);
    it += __shfl_xor(it, off, 32);
    ik += __shfl_xor(ik, off, 32);
  }
  if (b == 0){ out[0] = l; out[1] = lt; out[2] = lk; out[3] = it; out[4] = ik; }
}

extern "C" void kernel_launch(void* const* d_in, const int* in_sizes, int n_in,
                              void* d_out, int out_size, void* d_ws, size_t ws_size,
                              hipStream_t stream){
  const float* maps = (const float*)d_in[0];   // (16,7,160,160)
  const float* gt   = (const float*)d_in[1];   // (16,640,640)
  const float* gtk  = (const float*)d_in[2];   // (16,6,640,640)
  const float* tm   = (const float*)d_in[3];   // (16,640,640)
  unsigned* ws = (unsigned*)d_ws;
  float* out = (float*)d_out;

  kz<<<(WS_TOTAL + 255u)/256u, 256, 0, stream>>>(ws);
  k0_hist_hi<<<dim3(Hc, B_), 256, 0, stream>>>(maps, gt, tm, ws);
  k1_select_hi<<<B_, 256, 0, stream>>>(ws);
  k2_hist_lo<<<dim3(Hc, B_), 256, 0, stream>>>(maps, gt, ws);
  k3_select_lo<<<B_, 256, 0, stream>>>(ws);
  k4_main<<<dim3(2, Hc, B_), 256, 0, stream>>>(maps, gt, gtk, tm, ws);
  k5_final<<<1, 32, 0, stream>>>(ws, out);
}